// GC_LSTM_71648644432463
// MI455X (gfx1250) — compile-verified
//
#include <hip/hip_runtime.h>
#include <hip/hip_bf16.h>
#include <cstdint>
#include <cstddef>

// Problem constants (match reference)
#define BATCH 32
#define NNODE 4096
#define HISTT 16
#define PREDT 8
#define NEDGE 65536
// IN_DIM=10, GCN_OUT=2, HID=32, 4H=128, K = 10+2+32 = 44

typedef float v2f __attribute__((ext_vector_type(2)));
typedef float v8f __attribute__((ext_vector_type(8)));

__device__ __forceinline__ float sigm(float x) { return 1.f / (1.f + __expf(-x)); }

__device__ __forceinline__ void atomAddF(float* p, float v) {
  __hip_atomic_fetch_add(p, v, __ATOMIC_RELAXED, __HIP_MEMORY_SCOPE_AGENT);
}

// ---------------------------------------------------------------------------
// Small utility kernels
// ---------------------------------------------------------------------------
__global__ void k_zero(float* __restrict__ p, int n) {
  int i = blockIdx.x * 256 + threadIdx.x;
  if (i < n) p[i] = 0.f;
}

__global__ void k_deg(const int* __restrict__ ei, float* __restrict__ deg) {
  int e = blockIdx.x * 256 + threadIdx.x;
  if (e < NEDGE) atomAddF(deg + ei[e], 1.f);  // count src occurrences
}

__global__ void k_dis(const float* __restrict__ deg, float* __restrict__ dis) {
  int n = blockIdx.x * 256 + threadIdx.x;
  if (n < NNODE) {
    float d = deg[n];
    dis[n] = d > 0.f ? rsqrtf(d) : 0.f;
  }
}

__global__ void k_enorm(const int* __restrict__ ei, const float* __restrict__ dis,
                        float* __restrict__ enorm) {
  int e = blockIdx.x * 256 + threadIdx.x;
  if (e < NEDGE) enorm[e] = -dis[ei[e]] * dis[ei[NEDGE + e]];
}

// ---------------------------------------------------------------------------
// Cheb conv, history phase:  p = x@W1, q = x@W0 + b  for all (b,t,n)
// ---------------------------------------------------------------------------
__global__ void k_proj_hist(const float* __restrict__ pm25, const float* __restrict__ feat,
                            const float* __restrict__ cheb_w, const float* __restrict__ cheb_b,
                            float* __restrict__ p, float* __restrict__ q) {
  int i = blockIdx.x * 256 + threadIdx.x;  // i = (b*16+t)*4096 + n
  if (i >= BATCH * HISTT * NNODE) return;
  int bt = i >> 12, b = bt >> 4, t = bt & 15, n = i & 4095;
  float x[10];
  x[0] = pm25[i];
  const float* fp = feat + (((size_t)b * 24 + t) * NNODE + n) * 9;
#pragma unroll
  for (int j = 0; j < 9; ++j) x[1 + j] = fp[j];
  float q0 = cheb_b[0], q1 = cheb_b[1], p0 = 0.f, p1 = 0.f;
#pragma unroll
  for (int f = 0; f < 10; ++f) {
    q0 += x[f] * cheb_w[f * 2 + 0];       // W0
    q1 += x[f] * cheb_w[f * 2 + 1];
    p0 += x[f] * cheb_w[20 + f * 2 + 0];  // W1
    p1 += x[f] * cheb_w[20 + f * 2 + 1];
  }
  size_t o = (size_t)i * 2;
  p[o] = p0; p[o + 1] = p1; q[o] = q0; q[o + 1] = q1;
}

// scatter: acc[bt,dst] += p[bt,src] * norm(e), all (bt,e)
__global__ void k_scatter_hist(const int* __restrict__ ei, const float* __restrict__ enorm,
                               const float* __restrict__ p, float* __restrict__ acc) {
  int idx = blockIdx.x * 256 + threadIdx.x;
  if (idx >= BATCH * HISTT * NEDGE) return;
  int bt = idx >> 16, e = idx & 65535;
  float nv = enorm[e];
  int s = ei[e], d = ei[NEDGE + e];
  size_t sb = ((size_t)bt * NNODE + s) * 2;
  size_t db = ((size_t)bt * NNODE + d) * 2;
  atomAddF(&acc[db + 0], p[sb + 0] * nv);
  atomAddF(&acc[db + 1], p[sb + 1] * nv);
}

// xg = sigmoid(q + acc), written in place into acc
__global__ void k_fin_hist(const float* __restrict__ q, float* __restrict__ acc) {
  int i = blockIdx.x * 256 + threadIdx.x;
  if (i < BATCH * HISTT * NNODE * 2) acc[i] = sigm(q[i] + acc[i]);
}

// ---------------------------------------------------------------------------
// Cheb conv, prediction phase (per step k): build x=[xn, feat], project, zero acc
// ---------------------------------------------------------------------------
__global__ void k_proj_pred(const float* __restrict__ xn, const float* __restrict__ feat,
                            const float* __restrict__ cheb_w, const float* __restrict__ cheb_b,
                            float* __restrict__ p, float* __restrict__ q, float* __restrict__ acc,
                            int kstep) {
  int i = blockIdx.x * 256 + threadIdx.x;  // i = b*4096 + n
  if (i >= BATCH * NNODE) return;
  int b = i >> 12, n = i & 4095;
  float x[10];
  x[0] = xn[i];
  const float* fp = feat + (((size_t)b * 24 + HISTT + kstep) * NNODE + n) * 9;
#pragma unroll
  for (int j = 0; j < 9; ++j) x[1 + j] = fp[j];
  float q0 = cheb_b[0], q1 = cheb_b[1], p0 = 0.f, p1 = 0.f;
#pragma unroll
  for (int f = 0; f < 10; ++f) {
    q0 += x[f] * cheb_w[f * 2 + 0];
    q1 += x[f] * cheb_w[f * 2 + 1];
    p0 += x[f] * cheb_w[20 + f * 2 + 0];
    p1 += x[f] * cheb_w[20 + f * 2 + 1];
  }
  size_t o = (size_t)i * 2;
  p[o] = p0; p[o + 1] = p1; q[o] = q0; q[o + 1] = q1;
  acc[o] = 0.f; acc[o + 1] = 0.f;  // pre-zero scatter target (stream-ordered)
}

__global__ void k_scatter_pred(const int* __restrict__ ei, const float* __restrict__ enorm,
                               const float* __restrict__ p, float* __restrict__ acc) {
  int idx = blockIdx.x * 256 + threadIdx.x;
  if (idx >= BATCH * NEDGE) return;
  int b = idx >> 16, e = idx & 65535;
  float nv = enorm[e];
  int s = ei[e], d = ei[NEDGE + e];
  size_t sb = ((size_t)b * NNODE + s) * 2;
  size_t db = ((size_t)b * NNODE + d) * 2;
  atomAddF(&acc[db + 0], p[sb + 0] * nv);
  atomAddF(&acc[db + 1], p[sb + 1] * nv);
}

// ---------------------------------------------------------------------------
// WMMA LSTM core. One wave = 16 rows (b,n pairs). Gate GEMM:
//   G[16x128] = Xcat[16x12] @ Wx[12x128] + H[16x32] @ Wh[32x128] + bias
// via V_WMMA_F32_16X16X4_F32: 3 + 8 K-steps per 16-col tile, 8 col tiles.
//
// fp32 16x16x4 layouts (wave32):
//  A[16x4]: lane L (m=L&15, hi=L>>4): {A[m][2hi], A[m][2hi+1]}   (b64 LDS load)
//  B[4x16]: lane L (n=L&15):          {B[2hi][n], B[2hi+1][n]}   -> W^T storage
//  C/D[16x16]: lane L, elem r:        D[r+8hi][16ct + (L&15)]
// ---------------------------------------------------------------------------
#define WT_STRIDE 48  // padded row length of W^T[128][48] (K=44)

__device__ __forceinline__ void load_weights(int tid,
    const float* __restrict__ x2h_w, const float* __restrict__ x2h_b,
    const float* __restrict__ h2h_w, const float* __restrict__ h2h_b,
    const float* __restrict__ fc_w,
    float* sWT, float* sBias, float* sFc) {
  for (int i = tid; i < 44 * 128; i += 256) {
    int k = i >> 7, n = i & 127;
    sWT[n * WT_STRIDE + k] = (k < 12) ? x2h_w[k * 128 + n] : h2h_w[(k - 12) * 128 + n];
  }
  if (tid < 128) sBias[tid] = x2h_b[tid] + h2h_b[tid];
  if (tid < 32) sFc[tid] = fc_w[tid];
}

__device__ __forceinline__ void gate_compute(const float* __restrict__ sXw,
                                             const float* __restrict__ sHw,
                                             const float* __restrict__ sWT,
                                             const float* __restrict__ sBias,
                                             int m, int hi, v8f* acc) {
  v2f ax[3], ah[8];
#pragma unroll
  for (int kk = 0; kk < 3; ++kk) ax[kk] = *(const v2f*)(sXw + m * 12 + 4 * kk + 2 * hi);
#pragma unroll
  for (int kk = 0; kk < 8; ++kk) ah[kk] = *(const v2f*)(sHw + m * 32 + 4 * kk + 2 * hi);
#pragma unroll
  for (int ct = 0; ct < 8; ++ct) {
    float bb = sBias[ct * 16 + m];
    v8f a = {bb, bb, bb, bb, bb, bb, bb, bb};
    const float* wcol = sWT + (ct * 16 + m) * WT_STRIDE;
#pragma unroll
    for (int kk = 0; kk < 3; ++kk) {  // x-part, K rows 0..11
      v2f bf = *(const v2f*)(wcol + 4 * kk + 2 * hi);
      a = __builtin_amdgcn_wmma_f32_16x16x4_f32(false, ax[kk], false, bf, (short)0, a,
                                                false, false);
    }
#pragma unroll
    for (int kk = 0; kk < 8; ++kk) {  // h-part, K rows 12..43
      v2f bf = *(const v2f*)(wcol + 12 + 4 * kk + 2 * hi);
      a = __builtin_amdgcn_wmma_f32_16x16x4_f32(false, ah[kk], false, bf, (short)0, a,
                                                false, false);
    }
    acc[ct] = a;
  }
}

// gate order: i = cols 0..31 (tiles 0,1) | f = 32..63 (2,3) | g = 64..95 (4,5) | o = 96..127 (6,7)
__device__ __forceinline__ void lstm_update(const v8f* acc, float* c01, float* h01,
                                            float* sHw, int m, int hi) {
#pragma unroll
  for (int ct = 0; ct < 2; ++ct) {
#pragma unroll
    for (int r = 0; r < 8; ++r) {
      float iv = sigm(acc[ct][r]);
      float fv = sigm(acc[2 + ct][r]);
      float gv = tanhf(acc[4 + ct][r]);
      float ov = sigm(acc[6 + ct][r]);
      float cn = fv * c01[ct * 8 + r] + iv * gv;
      float hn = ov * tanhf(cn);
      c01[ct * 8 + r] = cn;
      h01[ct * 8 + r] = hn;
      sHw[(r + 8 * hi) * 32 + ct * 16 + m] = hn;  // D layout -> row-major H
    }
  }
}

// ---------------------------------------------------------------------------
// History phase: 16 LSTM steps with resident state; epilogue computes xn=h@fc+b
// ---------------------------------------------------------------------------
__launch_bounds__(256)
__global__ void hist_lstm_kernel(
    const float* __restrict__ pm25, const float* __restrict__ feat,
    const float* __restrict__ xg,
    const float* __restrict__ x2h_w, const float* __restrict__ x2h_b,
    const float* __restrict__ h2h_w, const float* __restrict__ h2h_b,
    const float* __restrict__ fc_w, const float* __restrict__ fc_b,
    float* __restrict__ h_glob, float* __restrict__ c_glob, float* __restrict__ xn_glob) {
  __shared__ float sWT[128 * WT_STRIDE];
  __shared__ float sBias[128];
  __shared__ float sFc[32];
  __shared__ float sX[8][16 * 12];
  __shared__ float sH[8][16 * 32];

  const int tid = threadIdx.x;
  load_weights(tid, x2h_w, x2h_b, h2h_w, h2h_b, fc_w, sWT, sBias, sFc);

  const int wave = tid >> 5, lane = tid & 31, m = lane & 15, hi = lane >> 4;
  const int rowbase = blockIdx.x * 128 + wave * 16;
  float* sXw = sX[wave];
  float* sHw = sH[wave];
  for (int j = lane; j < 512; j += 32) sHw[j] = 0.f;
  float c01[16], h01[16];
#pragma unroll
  for (int j = 0; j < 16; ++j) { c01[j] = 0.f; h01[j] = 0.f; }
  __syncthreads();

  for (int t = 0; t < HISTT; ++t) {
    if (lane < 16) {  // assemble Xcat = [pm25, feat(9), xg(2)] for this wave's 16 rows
      int R = rowbase + lane;
      int b = R >> 12, n = R & 4095;
      size_t hb = ((size_t)(b * 16 + t)) * NNODE + n;
      sXw[lane * 12 + 0] = pm25[hb];
      const float* fp = feat + (((size_t)b * 24 + t) * NNODE + n) * 9;
#pragma unroll
      for (int j = 0; j < 9; ++j) sXw[lane * 12 + 1 + j] = fp[j];
      sXw[lane * 12 + 10] = xg[hb * 2 + 0];
      sXw[lane * 12 + 11] = xg[hb * 2 + 1];
    }
    __syncthreads();
    v8f acc[8];
    gate_compute(sXw, sHw, sWT, sBias, m, hi, acc);
    lstm_update(acc, c01, h01, sHw, m, hi);
  }

  // store h, c state (D layout scatter)
#pragma unroll
  for (int ct = 0; ct < 2; ++ct)
#pragma unroll
    for (int r = 0; r < 8; ++r) {
      size_t R = (size_t)rowbase + r + 8 * hi;
      c_glob[R * 32 + ct * 16 + m] = c01[ct * 8 + r];
      h_glob[R * 32 + ct * 16 + m] = h01[ct * 8 + r];
    }
  __syncthreads();
  if (lane < 16) {  // xn = h @ fc_w + fc_b
    float s = fc_b[0];
#pragma unroll
    for (int j = 0; j < 32; ++j) s += sHw[lane * 32 + j] * sFc[j];
    xn_glob[rowbase + lane] = s;
  }
}

// ---------------------------------------------------------------------------
// Prediction phase: one LSTM step; reload state, finish cheb (sigmoid(q+acc)),
// write new state + xn + output
// ---------------------------------------------------------------------------
__launch_bounds__(256)
__global__ void pred_lstm_kernel(
    const float* __restrict__ feat,
    const float* __restrict__ q_pred, const float* __restrict__ acc_pred,
    const float* __restrict__ x2h_w, const float* __restrict__ x2h_b,
    const float* __restrict__ h2h_w, const float* __restrict__ h2h_b,
    const float* __restrict__ fc_w, const float* __restrict__ fc_b,
    float* __restrict__ h_glob, float* __restrict__ c_glob, float* __restrict__ xn_glob,
    float* __restrict__ out, int kstep) {
  __shared__ float sWT[128 * WT_STRIDE];
  __shared__ float sBias[128];
  __shared__ float sFc[32];
  __shared__ float sX[8][16 * 12];
  __shared__ float sH[8][16 * 32];

  const int tid = threadIdx.x;
  load_weights(tid, x2h_w, x2h_b, h2h_w, h2h_b, fc_w, sWT, sBias, sFc);

  const int wave = tid >> 5, lane = tid & 31, m = lane & 15, hi = lane >> 4;
  const int rowbase = blockIdx.x * 128 + wave * 16;
  float* sXw = sX[wave];
  float* sHw = sH[wave];
  for (int j = lane; j < 512; j += 32) sHw[j] = h_glob[(size_t)rowbase * 32 + j];
  float c01[16], h01[16];
#pragma unroll
  for (int ct = 0; ct < 2; ++ct)
#pragma unroll
    for (int r = 0; r < 8; ++r)
      c01[ct * 8 + r] = c_glob[((size_t)rowbase + r + 8 * hi) * 32 + ct * 16 + m];

  if (lane < 16) {
    int R = rowbase + lane;
    int b = R >> 12, n = R & 4095;
    sXw[lane * 12 + 0] = xn_glob[R];
    const float* fp = feat + (((size_t)b * 24 + HISTT + kstep) * NNODE + n) * 9;
#pragma unroll
    for (int j = 0; j < 9; ++j) sXw[lane * 12 + 1 + j] = fp[j];
    sXw[lane * 12 + 10] = sigm(q_pred[(size_t)R * 2 + 0] + acc_pred[(size_t)R * 2 + 0]);
    sXw[lane * 12 + 11] = sigm(q_pred[(size_t)R * 2 + 1] + acc_pred[(size_t)R * 2 + 1]);
  }
  __syncthreads();

  v8f acc[8];
  gate_compute(sXw, sHw, sWT, sBias, m, hi, acc);
  lstm_update(acc, c01, h01, sHw, m, hi);

#pragma unroll
  for (int ct = 0; ct < 2; ++ct)
#pragma unroll
    for (int r = 0; r < 8; ++r) {
      size_t R = (size_t)rowbase + r + 8 * hi;
      c_glob[R * 32 + ct * 16 + m] = c01[ct * 8 + r];
      h_glob[R * 32 + ct * 16 + m] = h01[ct * 8 + r];
    }
  __syncthreads();
  if (lane < 16) {
    float s = fc_b[0];
#pragma unroll
    for (int j = 0; j < 32; ++j) s += sHw[lane * 32 + j] * sFc[j];
    int R = rowbase + lane;
    int b = R >> 12, n = R & 4095;
    xn_glob[R] = s;
    out[((size_t)b * PREDT + kstep) * NNODE + n] = s;  // [B, PRED, N]
  }
}

// ---------------------------------------------------------------------------
// Launch
// ---------------------------------------------------------------------------
extern "C" void kernel_launch(void* const* d_in, const int* in_sizes, int n_in,
                              void* d_out, int out_size, void* d_ws, size_t ws_size,
                              hipStream_t stream) {
  (void)in_sizes; (void)n_in; (void)out_size; (void)ws_size;
  const float* pm25   = (const float*)d_in[0];
  const float* feat   = (const float*)d_in[1];
  const int*   ei     = (const int*)d_in[2];
  const float* cheb_w = (const float*)d_in[3];
  const float* cheb_b = (const float*)d_in[4];
  const float* x2h_w  = (const float*)d_in[5];
  const float* x2h_b  = (const float*)d_in[6];
  const float* h2h_w  = (const float*)d_in[7];
  const float* h2h_b  = (const float*)d_in[8];
  const float* fc_w   = (const float*)d_in[9];
  const float* fc_b   = (const float*)d_in[10];
  float* out = (float*)d_out;
  char* ws = (char*)d_ws;

  // workspace carve-out (bytes)
  float* deg      = (float*)(ws + 0);           //  16 KB (N)
  float* dis      = (float*)(ws + 0x4000);      //  16 KB (N)
  float* enorm    = (float*)(ws + 0x8000);      // 256 KB (E)
  float* p_hist   = (float*)(ws + 0x100000);    //  16 MB (B*T*N*2)
  float* q_hist   = (float*)(ws + 0x1100000);   //  16 MB
  float* acc_hist = (float*)(ws + 0x2100000);   //  16 MB (becomes xg)
  float* xn_glob  = (float*)(ws + 0x3100000);   // 512 KB (B*N)
  float* p_pred   = (float*)(ws + 0x3180000);   //   1 MB (B*N*2)
  float* q_pred   = (float*)(ws + 0x3280000);   //   1 MB
  float* acc_pred = (float*)(ws + 0x3380000);   //   1 MB   (total ~55 MB)
  float* h_glob = p_hist;  // p_hist/q_hist dead after k_fin_hist -> reuse for state
  float* c_glob = q_hist;

  // graph normalization (once)
  k_zero<<<(NNODE + 255) / 256, 256, 0, stream>>>(deg, NNODE);
  k_zero<<<(BATCH * HISTT * NNODE * 2 + 255) / 256, 256, 0, stream>>>(
      acc_hist, BATCH * HISTT * NNODE * 2);
  k_deg<<<NEDGE / 256, 256, 0, stream>>>(ei, deg);
  k_dis<<<NNODE / 256, 256, 0, stream>>>(deg, dis);
  k_enorm<<<NEDGE / 256, 256, 0, stream>>>(ei, dis, enorm);

  // history-phase cheb conv for all 16 steps in parallel
  k_proj_hist<<<BATCH * HISTT * NNODE / 256, 256, 0, stream>>>(pm25, feat, cheb_w, cheb_b,
                                                               p_hist, q_hist);
  k_scatter_hist<<<BATCH * HISTT * NEDGE / 256, 256, 0, stream>>>(ei, enorm, p_hist, acc_hist);
  k_fin_hist<<<BATCH * HISTT * NNODE * 2 / 256, 256, 0, stream>>>(q_hist, acc_hist);

  // 16-step resident-state WMMA LSTM
  hist_lstm_kernel<<<BATCH * NNODE / 128, 256, 0, stream>>>(
      pm25, feat, acc_hist, x2h_w, x2h_b, h2h_w, h2h_b, fc_w, fc_b, h_glob, c_glob, xn_glob);

  // 8 prediction steps (grid-wide sync via stream ordering)
  for (int k = 0; k < PREDT; ++k) {
    k_proj_pred<<<BATCH * NNODE / 256, 256, 0, stream>>>(xn_glob, feat, cheb_w, cheb_b,
                                                         p_pred, q_pred, acc_pred, k);
    k_scatter_pred<<<BATCH * NEDGE / 256, 256, 0, stream>>>(ei, enorm, p_pred, acc_pred);
    pred_lstm_kernel<<<BATCH * NNODE / 128, 256, 0, stream>>>(
        feat, q_pred, acc_pred, x2h_w, x2h_b, h2h_w, h2h_b, fc_w, fc_b,
        h_glob, c_glob, xn_glob, out, k);
  }
}